// VariogramScore_74277164417458
// MI455X (gfx1250) — compile-verified
//
#include <hip/hip_runtime.h>
#include <math.h>

// ---------------------------------------------------------------------------
// VariogramScore on MI455X (gfx1250, wave32).
//
// (x_i - x_j)^4 = sum_k C(4,k)(-1)^k x_i^{4-k} x_j^k  =>  per feature v,
// the (td - pd) L x L matrix is a rank-40 GEMM:
//   A[i,(e,k)]      =  c_k/E * t_{e,i}^{4-k}   (cols  0..19)
//   A[i,20+(e,k)]   = -c_k/E * p_{e,i}^{4-k}   (cols 20..39)
//   B[(e,k),j]      =  t_{e,j}^k ;  B[20+..,j] = p_{e,j}^k
// M = A*B is accumulated per 16x16 tile with 10 chained
// V_WMMA_F32_16X16X4_F32 ops (K=40), then squared+reduced in-register.
// ---------------------------------------------------------------------------

typedef float v2f __attribute__((ext_vector_type(2)));
typedef float v8f __attribute__((ext_vector_type(8)));

#define LDIM   1024
#define VDIM   16
#define EDIM   4
#define KDIM   40          // 2 (t,p) * E * 5 binomial terms
#define TPD    64          // 16x16 tiles per L dimension
#define NTILE  (TPD * TPD * VDIM)   // 65536

// -------------------- k0: sum(node_weights) ---------------------------------
__global__ __launch_bounds__(1024)
void vgs_k0_nwsum(const float* __restrict__ nw, float* __restrict__ nwsum) {
    __shared__ float red[1024];
    const int t = threadIdx.x;
    red[t] = nw[t];
    __syncthreads();
    for (int s = 512; s > 0; s >>= 1) {
        if (t < s) red[t] += red[t + s];
        __syncthreads();
    }
    if (t == 0) nwsum[0] = red[0];
}

// -------------------- k1: build powered WMMA operands -----------------------
// Layout: rows of 40 contiguous floats, [v][l][k]. Row stride 40 floats
// (160 B) keeps every (K, K+1) lane pair 8-byte aligned -> global_load_b64.
__global__ __launch_bounds__(256)
void vgs_k1_build(const float* __restrict__ preds,
                  const float* __restrict__ target,
                  const float* __restrict__ nw,
                  const float* __restrict__ nwsum,
                  float* __restrict__ Abuf,
                  float* __restrict__ Bbuf) {
    const int gid = blockIdx.x * 256 + threadIdx.x;     // 16384 threads
    const int l = gid & (LDIM - 1);
    const int v = gid >> 10;
    // nw_norm = (nw/sum)^(1/(BETA*2)) = (nw/sum)^(1/4)
    const float w = sqrtf(sqrtf(nw[l] / nwsum[0]));
    // c_k * (1/E) = {1,-4,6,-4,1} * 0.25
    const float c4[5] = {0.25f, -1.0f, 1.5f, -1.0f, 0.25f};

    float* Ar = Abuf + ((size_t)((v << 10) + l)) * KDIM;
    float* Br = Bbuf + ((size_t)((v << 10) + l)) * KDIM;

#pragma unroll
    for (int e = 0; e < EDIM; ++e) {
        const int src = (e * LDIM + l) * VDIM + v;
        const float tv = target[src] * w;
        const float pv = preds[src]  * w;
        float tp[5], pp[5];
        tp[0] = 1.0f; pp[0] = 1.0f;
#pragma unroll
        for (int k = 1; k < 5; ++k) { tp[k] = tp[k-1] * tv; pp[k] = pp[k-1] * pv; }
#pragma unroll
        for (int k = 0; k < 5; ++k) {
            Ar[e * 5 + k]      =  c4[k] * tp[4 - k];   // td contribution
            Br[e * 5 + k]      =  tp[k];
            Ar[20 + e * 5 + k] = -c4[k] * pp[4 - k];   // -pd contribution
            Br[20 + e * 5 + k] =  pp[k];
        }
    }
}

// -------------------- k2: WMMA tile kernel ----------------------------------
// One wave32 per 16x16 output tile; 8 waves / 256-thread block.
// f32 16x16x4 operand layout (ISA 7.12.2): lane = 16*half + m,
//   A VGPR0 <- K = 2*half, VGPR1 <- K = 2*half+1  (rows m, K-pairs contiguous)
//   B mirrored with N = lane%16. Both operands stored K-contiguous so each
//   lane fetches its (K,K+1) pair with a single 8-byte load.
__global__ __launch_bounds__(256)
void vgs_k2_tiles(const float* __restrict__ Abuf,
                  const float* __restrict__ Bbuf,
                  const float* __restrict__ fw,
                  float* __restrict__ partials) {
    const int lane = threadIdx.x & 31;
    const int wv   = threadIdx.x >> 5;
    const int tile = blockIdx.x * 8 + wv;               // 0..65535
    const int v    = tile >> 12;                        // 4096 tiles per v
    const int rem  = tile & 4095;
    const int ti   = rem >> 6;
    const int tj   = rem & 63;
    const int m    = lane & 15;
    const int h    = lane >> 4;

    const float* Ar = Abuf + ((size_t)((v << 10) + ti * 16 + m)) * KDIM + 2 * h;
    const float* Br = Bbuf + ((size_t)((v << 10) + tj * 16 + m)) * KDIM + 2 * h;

    v8f acc = {};
#pragma unroll
    for (int kk = 0; kk < 10; ++kk) {                   // K = 40, step 4
        const v2f a = *(const v2f*)(Ar + kk * 4);
        const v2f b = *(const v2f*)(Br + kk * 4);
        // (neg_a, A, neg_b, B, c_mod, C, reuse_a, reuse_b)
        acc = __builtin_amdgcn_wmma_f32_16x16x4_f32(
            false, a, false, b, (short)0, acc, false, false);
    }

    // acc now holds this wave's 8 elements of (td - pd); square + reduce.
    float s = 0.0f;
#pragma unroll
    for (int i = 0; i < 8; ++i) s += acc[i] * acc[i];
#pragma unroll
    for (int off = 16; off > 0; off >>= 1) s += __shfl_xor(s, off, 32);

    // final = sum_v fw[v]/256 * S_v ; fold the weight in per tile.
    if (lane == 0) partials[tile] = s * fw[v] * (1.0f / 256.0f);
}

// -------------------- k3: deterministic final reduction ---------------------
__global__ __launch_bounds__(256)
void vgs_k3_reduce(const float* __restrict__ partials, float* __restrict__ out) {
    __shared__ float red[256];
    const int t = threadIdx.x;
    float s = 0.0f;
    for (int i = t; i < NTILE; i += 256) s += partials[i];  // fixed order
    red[t] = s;
    __syncthreads();
    for (int st = 128; st > 0; st >>= 1) {
        if (t < st) red[t] += red[t + st];
        __syncthreads();
    }
    if (t == 0) out[0] = red[0];
}

// ---------------------------------------------------------------------------
extern "C" void kernel_launch(void* const* d_in, const int* in_sizes, int n_in,
                              void* d_out, int out_size, void* d_ws, size_t ws_size,
                              hipStream_t stream) {
    const float* preds  = (const float*)d_in[0];   // (1,4,1,1024,16)
    const float* target = (const float*)d_in[1];
    const float* nw     = (const float*)d_in[2];   // (1024,)
    const float* fw     = (const float*)d_in[3];   // (16,)
    float* out = (float*)d_out;

    float* ws       = (float*)d_ws;
    float* nwsum    = ws;                               // 64-float pad
    float* Abuf     = ws + 64;                          // 16*1024*40 floats
    float* Bbuf     = Abuf + (size_t)VDIM * LDIM * KDIM;
    float* partials = Bbuf + (size_t)VDIM * LDIM * KDIM; // 65536 floats
    (void)in_sizes; (void)n_in; (void)out_size; (void)ws_size;

    vgs_k0_nwsum <<<1, 1024, 0, stream>>>(nw, nwsum);
    vgs_k1_build <<<(VDIM * LDIM) / 256, 256, 0, stream>>>(preds, target, nw,
                                                           nwsum, Abuf, Bbuf);
    vgs_k2_tiles <<<NTILE / 8, 256, 0, stream>>>(Abuf, Bbuf, fw, partials);
    vgs_k3_reduce<<<1, 256, 0, stream>>>(partials, out);
}